// SpikingEmbed_51762945852120
// MI455X (gfx1250) — compile-verified
//
#include <hip/hip_runtime.h>

// ---------------------------------------------------------------------------
// SpikingEmbed forward for MI455X (gfx1250, wave32, WMMA)
//   conv1d(64->256,k=3,SAME) as implicit GEMM with V_WMMA_F32_16X16X4_F32
//   -> y[T,B,C] (fp32, staged in ws; 144 MB working set is L2-resident, 192 MB L2)
//   -> deterministic per-channel BN stats -> folded scale/shift
//   -> segmented LIF scan (tau=2 decay => 64-step warmup, 131072-way parallel)
// ---------------------------------------------------------------------------

typedef __attribute__((ext_vector_type(2))) float v2f;
typedef __attribute__((ext_vector_type(8))) float v8f;

#define B_    16
#define T_    4096
#define CIN   64
#define C_    256
#define KTOT  192            // CIN * 3
#define ROWS  (T_ * B_)      // 65536 rows of y[T,B,C]
#define SEG   128
#define WARM  64
#define NSTATBLK 256         // blocks in BN partial reduction

// --- repack conv_w [C, CIN, 3] -> Bprep pairs: v2f[(k>>1)*256 + n] = (B[k][n], B[k+1][n])
// with k = kt*64 + ci  (a k-quad starting at even K0 loads one contiguous float2)
__global__ void __launch_bounds__(256)
wprep_kernel(const float* __restrict__ w, float* __restrict__ bprep) {
    int idx = blockIdx.x * 256 + threadIdx.x;          // 0 .. 49151
    if (idx >= KTOT * C_) return;
    int k  = idx >> 8;                                 // 0..191
    int n  = idx & 255;                                // output channel
    int ci = k & 63;
    int kt = k >> 6;
    float v = w[n * (CIN * 3) + ci * 3 + kt];
    bprep[(((k >> 1) * C_) + n) * 2 + (k & 1)] = v;
}

// --- WMMA main loop; CHK=true only for the two boundary m-tiles per batch
template <bool CHK>
__device__ __forceinline__ void
conv_mainloop(const float* __restrict__ x, const v2f* __restrict__ bv,
              int b, int t0, int row, int half, int n0,
              v8f& acc0, v8f& acc1, v8f& acc2, v8f& acc3) {
#pragma unroll 4
    for (int K0 = 0; K0 < KTOT; K0 += 4) {
        const int kt = K0 >> 6;
        const int ci = (K0 & 63) + (half << 1);
        const int tt = t0 + row + kt - 1;               // local t within batch
        v2f a;
        if (CHK) {
            a = (v2f){0.0f, 0.0f};
            if (tt >= 0 && tt < T_)
                a = *(const v2f*)(x + ((size_t)(b * T_ + tt)) * CIN + ci);
        } else {
            a = *(const v2f*)(x + ((size_t)(b * T_ + tt)) * CIN + ci);
        }
        const int kk = (K0 >> 1) + half;                // pair index (k even)
        const v2f* bq = bv + (size_t)kk * C_ + n0 + row;
        v2f b0 = bq[0], b1 = bq[16], b2 = bq[32], b3 = bq[48];

        acc0 = __builtin_amdgcn_wmma_f32_16x16x4_f32(false, a, false, b0, (short)0, acc0, false, false);
        acc1 = __builtin_amdgcn_wmma_f32_16x16x4_f32(false, a, false, b1, (short)0, acc1, false, false);
        acc2 = __builtin_amdgcn_wmma_f32_16x16x4_f32(false, a, false, b2, (short)0, acc2, false, false);
        acc3 = __builtin_amdgcn_wmma_f32_16x16x4_f32(false, a, false, b3, (short)0, acc3, false, false);
    }
}

// --- conv as GEMM: one wave computes a 16(M) x 64(N) tile; M = b*T + t.
// A[m][k] = x[b, t + kt - 1, ci] (zero padded), B[k][n] = w[n][ci][kt].
__global__ void __launch_bounds__(256)
conv_wmma_kernel(const float* __restrict__ x, const float* __restrict__ bprep,
                 const float* __restrict__ bias, float* __restrict__ y) {
    const int lane  = threadIdx.x & 31;
    const int wv    = blockIdx.x * 8 + (threadIdx.x >> 5);  // 0..16383
    const int mTile = wv >> 2;                              // 0..4095
    const int n0    = (wv & 3) * 64;
    const int b     = mTile >> 8;                           // 256 m-tiles per batch
    const int t0    = (mTile & 255) << 4;
    const int row   = lane & 15;
    const int half  = lane >> 4;

    v8f acc0 = {}, acc1 = {}, acc2 = {}, acc3 = {};
    const v2f* bv = (const v2f*)bprep;

    if (t0 > 0 && t0 < (T_ - 16))
        conv_mainloop<false>(x, bv, b, t0, row, half, n0, acc0, acc1, acc2, acc3);
    else
        conv_mainloop<true >(x, bv, b, t0, row, half, n0, acc0, acc1, acc2, acc3);

    // epilogue: + bias, store to y[T,B,C]; D layout: VGPR j -> M=j (lanes 0-15) / j+8 (16-31)
    const float bi0 = bias[n0 +  0 + row];
    const float bi1 = bias[n0 + 16 + row];
    const float bi2 = bias[n0 + 32 + row];
    const float bi3 = bias[n0 + 48 + row];
    const int mbase = t0 + half * 8;
#pragma unroll
    for (int j = 0; j < 8; ++j) {
        size_t rbase = ((size_t)(mbase + j) * B_ + b) * C_;
        y[rbase + n0 +  0 + row] = acc0[j] + bi0;
        y[rbase + n0 + 16 + row] = acc1[j] + bi1;
        y[rbase + n0 + 32 + row] = acc2[j] + bi2;
        y[rbase + n0 + 48 + row] = acc3[j] + bi3;
    }
}

// --- deterministic per-channel partial sums: block i reduces 256 rows,
// writes partial[i*512 + c] = sum, partial[i*512 + 256 + c] = sumsq
__global__ void __launch_bounds__(256)
bnstat_kernel(const float* __restrict__ y, float* __restrict__ partial) {
    const int c = threadIdx.x;
    const float* p = y + (size_t)blockIdx.x * 256 * C_ + c;
    float s = 0.0f, q = 0.0f;
#pragma unroll 4
    for (int i = 0; i < 256; ++i) {
        float v = p[(size_t)i * C_];
        s += v; q += v * v;
    }
    partial[(size_t)blockIdx.x * 512 + c]       = s;
    partial[(size_t)blockIdx.x * 512 + 256 + c] = q;
}

// --- fixed-order final reduction + fold BN into per-channel affine
__global__ void __launch_bounds__(256)
bnfinal_kernel(const float* __restrict__ partial,
               const float* __restrict__ gamma, const float* __restrict__ beta,
               float* __restrict__ scale, float* __restrict__ shift) {
    const int c = threadIdx.x;
    float s = 0.0f, q = 0.0f;
#pragma unroll 4
    for (int i = 0; i < NSTATBLK; ++i) {
        s += partial[(size_t)i * 512 + c];
        q += partial[(size_t)i * 512 + 256 + c];
    }
    const float invN = 1.0f / (float)ROWS;
    float mean = s * invN;
    float var  = q * invN - mean * mean;
    float sc   = gamma[c] * rsqrtf(var + 1e-5f);
    scale[c] = sc;
    shift[c] = beta[c] - mean * sc;
}

// --- segmented LIF: 32 segments x 4096 (b,c) lanes; warmup from v=0 (tau=2
// decay + hard resets make the 64-step warmup numerically exact)
__global__ void __launch_bounds__(256)
lif_kernel(const float* __restrict__ y, const float* __restrict__ scale,
           const float* __restrict__ shift, float* __restrict__ out) {
    const int seg = blockIdx.x >> 4;                        // 0..31
    const int id  = (blockIdx.x & 15) * 256 + threadIdx.x;  // 0..4095 = b*256+c
    const int c   = id & 255;
    const float sc = scale[c], sh = shift[c];
    const int tstart = seg * SEG;
    int tw = tstart - WARM; if (tw < 0) tw = 0;

    float v = 0.0f;
    for (int t = tw; t < tstart; ++t) {
        float u = __builtin_fmaf(y[(size_t)t * 4096 + id], sc, sh);
        v = 0.5f * v + u;
        if (v >= 1.0f) v = 0.0f;
    }
#pragma unroll 4
    for (int t = tstart; t < tstart + SEG; ++t) {
        __builtin_prefetch(y + (size_t)(t + 8) * 4096 + id, 0, 0);
        float u = __builtin_fmaf(y[(size_t)t * 4096 + id], sc, sh);
        v = 0.5f * v + u;
        float s = (v >= 1.0f) ? 1.0f : 0.0f;
        out[(size_t)t * 4096 + id] = s;
        v = (v >= 1.0f) ? 0.0f : v;
    }
}

extern "C" void kernel_launch(void* const* d_in, const int* in_sizes, int n_in,
                              void* d_out, int out_size, void* d_ws, size_t ws_size,
                              hipStream_t stream) {
    const float* x      = (const float*)d_in[0];   // [16, 4096, 64]
    const float* conv_w = (const float*)d_in[1];   // [256, 64, 3]
    const float* conv_b = (const float*)d_in[2];   // [256]
    const float* gamma  = (const float*)d_in[3];   // [256]
    const float* beta   = (const float*)d_in[4];   // [256]
    float* out = (float*)d_out;                    // [4096, 16, 256]

    char* ws = (char*)d_ws;
    const size_t Y_BYTES      = (size_t)ROWS * C_ * 4;       // 64 MB
    const size_t BPREP_BYTES  = (size_t)KTOT * C_ * 4;       // 192 KB
    const size_t PART_BYTES   = (size_t)NSTATBLK * 512 * 4;  // 512 KB
    float* y       = (float*)(ws);
    float* bprep   = (float*)(ws + Y_BYTES);
    float* partial = (float*)(ws + Y_BYTES + BPREP_BYTES);
    float* scale   = (float*)(ws + Y_BYTES + BPREP_BYTES + PART_BYTES);
    float* shift   = scale + 256;

    wprep_kernel    <<<192,      256, 0, stream>>>(conv_w, bprep);
    conv_wmma_kernel<<<2048,     256, 0, stream>>>(x, bprep, conv_b, y);
    bnstat_kernel   <<<NSTATBLK, 256, 0, stream>>>(y, partial);
    bnfinal_kernel  <<<1,        256, 0, stream>>>(partial, gamma, beta, scale, shift);
    lif_kernel      <<<512,      256, 0, stream>>>(y, scale, shift, out);
}